// adjoint_patch_17386027614218
// MI455X (gfx1250) — compile-verified
//
#include <hip/hip_runtime.h>
#include <stdint.h>

// ---------------------------------------------------------------------------
// FBP adjoint: out[b,i,j,k] = (1/A) * sum_a bilerp(x[b,a], gx(a,i,k), gy=j)
// linspace steps are exact (0.0625) => gy collapses to row j exactly.
// Per j: Out_j[128x1089] = X_j[128 x 121*36] @ W[121*36 x 1089]
// via V_WMMA_F32_16X16X4_F32 (2x2 register blocking) with double-buffered
// GLOBAL_LOAD_ASYNC_TO_LDS staging (ASYNCcnt) overlapping DMA with compute.
// ---------------------------------------------------------------------------

typedef float v2f __attribute__((ext_vector_type(2)));
typedef float v8f __attribute__((ext_vector_type(8)));

#define B_DIM   128
#define A_DIM   121
#define P_DIM   33
#define N_DIM   (P_DIM * P_DIM)      // 1089 (i,k) pairs
#define KW      36                   // 33 w's padded to multiple of 4
#define NT      69                   // ceil(1089/16) N tiles
#define KC      9                    // KW/4 k-chunks per angle
#define AC      8                    // angles staged per LDS buffer
#define WFRAG   64                   // floats per packed B fragment (16x16x4)
#define MSUP    32                   // M rows per block (2 WMMA tiles)
#define BUFSZ   (AC * MSUP * KW)     // floats per LDS buffer (9216)
#define NCH     ((A_DIM + AC - 1) / AC)  // 16 angle chunks

// Packed W layout: Wpk[((a*NT + nt)*KC + kc)*64 + lane*2 + r]
//   lane l holds B elements (K = kc*4 + 2*(l>=16) + r, N = nt*16 + l%16)
#define W_TOTAL (A_DIM * NT * KC * WFRAG)   // 4,809,024 floats (~19.2 MB)

__global__ __launch_bounds__(256)
void fbp_build_w(const float* __restrict__ angles, float* __restrict__ Wpk) {
    int E = blockIdx.x * 256 + threadIdx.x;
    if (E >= W_TOTAL) return;
    int f    = E & 63;
    int kc   = (E >> 6) % KC;
    int nt   = (E / (64 * KC)) % NT;
    int a    = E / (64 * KC * NT);
    int lane = f >> 1;
    int r    = f & 1;
    int n    = nt * 16 + (lane & 15);           // (i,k) flat index
    int K    = kc * 4 + ((lane >> 4) << 1) + r; // w index (0..35)

    float val = 0.0f;
    if (n < N_DIM) {
        int i = n / P_DIM;
        int k = n % P_DIM;
        float ang = angles[a];
        float c = cosf(ang);
        float s = sinf(ang);
        float lin_i = -1.0f + 0.0625f * (float)i;   // exact linspace values
        float lin_k = -1.0f + 0.0625f * (float)k;
        float ix = (lin_k * c + lin_i * s + 1.0f) * 0.5f * 32.0f;
        float ix0f = floorf(ix);
        float fr   = ix - ix0f;
        int ix0    = (int)ix0f;
        if (K == ix0 && ix0 >= 0 && ix0 <= 32)                  val = 1.0f - fr;
        else if (K == ix0 + 1 && ix0 + 1 >= 0 && ix0 + 1 <= 32) val = fr;
        val *= (1.0f / (float)A_DIM);               // fold 1/A into weights
    }
    Wpk[E] = val;
}

// Issue async DMA of x[ms*32..+32, a0..a0+8, j, 0..32] into LDS buffer `sbuf`.
// 8448 dwords, 256 threads -> exactly 33 issues/thread (uniform, no divergence).
__device__ __forceinline__
void stage_async(const float* __restrict__ x, float* sbuf,
                 int ms, int j, int a0, int tid) {
    #pragma unroll 1
    for (int idx = tid; idx < AC * MSUP * P_DIM; idx += 256) {
        int w  = idx % P_DIM;
        int ba = idx / P_DIM;      // ac*32 + bb
        int bb = ba & 31;
        int ac = ba >> 5;
        int a  = a0 + ac;
        if (a > A_DIM - 1) a = A_DIM - 1;   // clamp (data unused, stay in-bounds)
        const float* gp = x + (((size_t)(ms * MSUP + bb) * A_DIM + (size_t)a) * P_DIM
                               + (size_t)j) * P_DIM + (size_t)w;
        uint32_t ldsoff = (uint32_t)(uintptr_t)(sbuf + (ba * KW + w));
        asm volatile("global_load_async_to_lds_b32 %0, %1, off"
                     :: "v"(ldsoff), "v"((uint64_t)(uintptr_t)gp)
                     : "memory");
    }
}

// grid: x = 5 N-chunks (8 waves * 2 tiles = 16 tiles/chunk, 5*16=80 >= 69),
//       y = j (33), z = M supertile (4, 32 rows each).
// block: 256 threads = 8 waves; wave wv owns N tiles {chunk*16+2wv, +1} and
//        both 16-row M tiles of the supertile (2x2 register blocking).
__global__ __launch_bounds__(256)
void fbp_wmma(const float* __restrict__ x,
              const float* __restrict__ Wpk,
              float* __restrict__ out) {
    const int j    = blockIdx.y;
    const int ms   = blockIdx.z;          // M supertile: rows ms*32 .. +32
    const int tid  = threadIdx.x;
    const int wv   = tid >> 5;
    const int lane = tid & 31;
    const int lr   = lane & 15;           // N within tile / b within tile
    const int hl   = lane >> 4;

    const int  nt0  = blockIdx.x * 16 + wv * 2;
    const int  nt1  = nt0 + 1;
    const bool v0   = (nt0 < NT);
    const bool v1   = (nt1 < NT);
    const int  ntc0 = v0 ? nt0 : (NT - 1);   // clamp: keep wave in loop
    const int  ntc1 = v1 ? nt1 : (NT - 1);

    __shared__ __align__(16) float sx[2 * BUFSZ];   // 73.7 KB double buffer

    v8f acc00 = {}, acc01 = {}, acc10 = {}, acc11 = {};

    const int a_base_lds = (lr * KW) + 2 * hl;  // per-lane A-frag base (floats)
    const float* wlane = Wpk + (size_t)(lane * 2);

    // Pre-zero pad columns w=33..35 of every row in BOTH buffers; async copies
    // never touch them, so they stay zero (weights for K>=33 are zero, but
    // 0 * uninitialized-NaN would poison the accumulators).
    for (int r = tid; r < 2 * AC * MSUP; r += 256) {
        int base = r * KW + P_DIM;
        sx[base] = 0.0f; sx[base + 1] = 0.0f; sx[base + 2] = 0.0f;
    }

    stage_async(x, sx, ms, j, 0, tid);          // prefetch chunk 0 -> buffer 0

    for (int c = 0; c < NCH; ++c) {
        asm volatile("s_wait_asynccnt 0x0" ::: "memory");  // my DMA done
        __syncthreads();   // all waves' DMA visible; prev compute finished

        if (c + 1 < NCH)   // overlap: DMA next chunk into the other buffer
            stage_async(x, sx + ((c + 1) & 1) * BUFSZ, ms, j, (c + 1) * AC, tid);

        const float* buf = sx + (c & 1) * BUFSZ;
        const int a0 = c * AC;
        const int na = (A_DIM - a0 < AC) ? (A_DIM - a0) : AC;
        for (int ac = 0; ac < na; ++ac) {
            const size_t abase = (size_t)(a0 + ac) * NT;
            const float* wrow0 = wlane + (abase + (size_t)ntc0) * (KC * WFRAG);
            const float* wrow1 = wlane + (abase + (size_t)ntc1) * (KC * WFRAG);
            const float* arow0 = buf + (ac * MSUP) * KW + a_base_lds;
            const float* arow1 = arow0 + 16 * KW;
            #pragma unroll
            for (int kc = 0; kc < KC; ++kc) {
                v2f af0 = *(const v2f*)(arow0 + kc * 4);      // ds_load_b64
                v2f af1 = *(const v2f*)(arow1 + kc * 4);      // ds_load_b64
                v2f bf0 = *(const v2f*)(wrow0 + kc * WFRAG);  // global_load_b64
                v2f bf1 = *(const v2f*)(wrow1 + kc * WFRAG);  // global_load_b64
                acc00 = __builtin_amdgcn_wmma_f32_16x16x4_f32(
                            false, af0, false, bf0, (short)0, acc00, false, false);
                acc01 = __builtin_amdgcn_wmma_f32_16x16x4_f32(
                            false, af0, false, bf1, (short)0, acc01, false, false);
                acc10 = __builtin_amdgcn_wmma_f32_16x16x4_f32(
                            false, af1, false, bf0, (short)0, acc10, false, false);
                acc11 = __builtin_amdgcn_wmma_f32_16x16x4_f32(
                            false, af1, false, bf1, (short)0, acc11, false, false);
            }
        }
    }

    // ---- store four 16x16 tiles ------------------------------------------
    // C/D layout: lane<16 -> rows M=v, lane>=16 -> rows M=v+8; N = lane%16.
    const int mrow0 = ms * MSUP + hl * 8;        // tile row base, M tile 0
    const int mrow1 = mrow0 + 16;                // M tile 1
    const int n0    = nt0 * 16 + lr;
    const int n1    = nt1 * 16 + lr;

    if (v0 && n0 < N_DIM) {
        int i = n0 / P_DIM, k = n0 % P_DIM;
        size_t obase = (size_t)i * (size_t)N_DIM + (size_t)j * P_DIM + (size_t)k;
        #pragma unroll
        for (int v = 0; v < 8; ++v) {
            out[(size_t)(mrow0 + v) * (size_t)(P_DIM * N_DIM) + obase] = acc00[v];
            out[(size_t)(mrow1 + v) * (size_t)(P_DIM * N_DIM) + obase] = acc10[v];
        }
    }
    if (v1 && n1 < N_DIM) {
        int i = n1 / P_DIM, k = n1 % P_DIM;
        size_t obase = (size_t)i * (size_t)N_DIM + (size_t)j * P_DIM + (size_t)k;
        #pragma unroll
        for (int v = 0; v < 8; ++v) {
            out[(size_t)(mrow0 + v) * (size_t)(P_DIM * N_DIM) + obase] = acc01[v];
            out[(size_t)(mrow1 + v) * (size_t)(P_DIM * N_DIM) + obase] = acc11[v];
        }
    }
}

extern "C" void kernel_launch(void* const* d_in, const int* in_sizes, int n_in,
                              void* d_out, int out_size, void* d_ws, size_t ws_size,
                              hipStream_t stream) {
    (void)in_sizes; (void)n_in; (void)out_size; (void)ws_size;
    const float* x      = (const float*)d_in[0];   // (128,121,33,33) f32
    const float* angles = (const float*)d_in[1];   // (1,121) f32
    float* out = (float*)d_out;                    // (128,33,33,33) f32
    float* Wpk = (float*)d_ws;                     // ~19.2 MB packed weights

    int wblocks = (W_TOTAL + 255) / 256;
    fbp_build_w<<<wblocks, 256, 0, stream>>>(angles, Wpk);

    dim3 grid(5, P_DIM, B_DIM / MSUP);  // N-chunks x j x M-supertiles
    fbp_wmma<<<grid, 256, 0, stream>>>(x, Wpk, out);
}